// Attn_20993800143223
// MI455X (gfx1250) — compile-verified
//
#include <hip/hip_runtime.h>

#define S_DIM 512
#define B_DIM 128
#define H_DIM 1024

typedef __attribute__((ext_vector_type(2))) float v2f;
typedef __attribute__((ext_vector_type(4))) float fvec4;
typedef __attribute__((ext_vector_type(8))) float v8f;

// ---------------------------------------------------------------------------
// Kernel 1: v[b,h] = sum_k hidden[b,k] * W[k,h]   (128x1024 = 8x64 tiles)
// One wave (32 lanes) per 16x16 output tile, K-loop in steps of 4 using
// V_WMMA_F32_16X16X4_F32.
// A (16x4, MxK) lane layout: lanes 0-15 -> M=lane, K=0/1 in the two VGPRs;
//                            lanes 16-31 -> M=lane-16, K=2/3.
// B (4x16, KxN) lane layout: lanes 0-15 -> N=lane, rows K=0/1 in the two
//                            VGPRs; lanes 16-31 -> rows K=2/3.
// D (16x16 f32): VGPR r -> M=r (lanes 0-15) / M=r+8 (lanes 16-31), N=lane&15.
// ---------------------------------------------------------------------------
__global__ __launch_bounds__(32) void gemm_v_kernel(const float* __restrict__ hidden,
                                                    const float* __restrict__ W,
                                                    float* __restrict__ v) {
    const int b0   = blockIdx.x * 16;   // batch tile   (grid.x = 8)
    const int n0   = blockIdx.y * 16;   // H-out tile   (grid.y = 64)
    const int lane = threadIdx.x;
    const int hi   = lane >> 4;         // 0 or 1
    const int l15  = lane & 15;
    const int kOff = hi * 2;

    const float* hRow = hidden + (size_t)(b0 + l15) * H_DIM + kOff;   // A source
    const float* wCol = W + (size_t)kOff * H_DIM + n0 + l15;          // B source

    v8f acc = {};
    for (int k = 0; k < H_DIM; k += 4) {
        v2f a = *(const v2f*)(hRow + k);            // A[m, k+kOff], A[m, k+kOff+1]
        v2f bm;
        bm.x = wCol[(size_t)(k + 0) * H_DIM];       // B[k+kOff  , n]
        bm.y = wCol[(size_t)(k + 1) * H_DIM];       // B[k+kOff+1, n]
        acc = __builtin_amdgcn_wmma_f32_16x16x4_f32(
            /*neg_a=*/false, a, /*neg_b=*/false, bm,
            /*c_mod=*/(short)0, acc, /*reuse_a=*/false, /*reuse_b=*/false);
    }

#pragma unroll
    for (int r = 0; r < 8; ++r) {
        const int m = r + hi * 8;
        v[(size_t)(b0 + m) * H_DIM + n0 + l15] = acc[r];
    }
}

// ---------------------------------------------------------------------------
// Kernel 2: c[b] = hidden[b,:] . bias   (one wave per b)
// ---------------------------------------------------------------------------
__global__ __launch_bounds__(256) void bias_dot_kernel(const float* __restrict__ hidden,
                                                       const float* __restrict__ bias,
                                                       float* __restrict__ c) {
    const int wave = threadIdx.x >> 5;
    const int lane = threadIdx.x & 31;
    const int b    = blockIdx.x * 8 + wave;

    float acc = 0.f;
    for (int h = lane; h < H_DIM; h += 32)
        acc += hidden[(size_t)b * H_DIM + h] * bias[h];
    for (int off = 16; off; off >>= 1)
        acc += __shfl_xor(acc, off, 32);
    if (lane == 0) c[b] = acc;
}

// ---------------------------------------------------------------------------
// Kernel 3: scores[b,s] = enc[s,b,:] . v[b,:] + c[b]
// Bandwidth-bound streaming of enc (268 MB). Grid (B, S/64), 256 threads.
// v[b,:] staged in LDS; per row each lane does 8x b128 NT loads (512 B
// contiguous per wave per load -> fully coalesced), then shuffle-reduce.
// ---------------------------------------------------------------------------
__global__ __launch_bounds__(256) void scores_kernel(const float* __restrict__ enc,
                                                     const float* __restrict__ v,
                                                     const float* __restrict__ c,
                                                     float* __restrict__ scores) {
    __shared__ fvec4 vs[H_DIM / 4];
    const int b      = blockIdx.x;
    const int s_base = blockIdx.y * 64;
    const int wave   = threadIdx.x >> 5;
    const int lane   = threadIdx.x & 31;

    // cooperative load of v[b,:] (4 KB) into LDS
    const fvec4* vsrc = (const fvec4*)(v + (size_t)b * H_DIM);
    for (int i = threadIdx.x; i < H_DIM / 4; i += 256)
        vs[i] = vsrc[i];
    __syncthreads();

    const float cb = c[b];

#pragma unroll 2
    for (int r = 0; r < 8; ++r) {
        const int s = s_base + wave * 8 + r;
        const fvec4* erow = (const fvec4*)(enc + ((size_t)s * B_DIM + b) * H_DIM);
        float acc = 0.f;
#pragma unroll
        for (int j = 0; j < 8; ++j) {
            fvec4 e = __builtin_nontemporal_load(erow + lane + j * 32);
            fvec4 w = vs[lane + j * 32];
            acc += e.x * w.x + e.y * w.y + e.z * w.z + e.w * w.w;
        }
        for (int off = 16; off; off >>= 1)
            acc += __shfl_xor(acc, off, 32);
        if (lane == 0)
            scores[(size_t)b * S_DIM + s] = acc + cb;
    }
}

// ---------------------------------------------------------------------------
// Kernel 4: softmax over s (512) per b; out[b,0,s]
// ---------------------------------------------------------------------------
__global__ __launch_bounds__(256) void softmax_kernel(const float* __restrict__ scores,
                                                      float* __restrict__ out) {
    __shared__ float red[8];
    const int b    = blockIdx.x;
    const int t    = threadIdx.x;
    const int wave = t >> 5;
    const int lane = t & 31;

    const float x0 = scores[(size_t)b * S_DIM + t];
    const float x1 = scores[(size_t)b * S_DIM + t + 256];

    // block max
    float m = fmaxf(x0, x1);
    for (int off = 16; off; off >>= 1)
        m = fmaxf(m, __shfl_xor(m, off, 32));
    if (lane == 0) red[wave] = m;
    __syncthreads();
    {
        float mm = red[lane & 7];
        for (int off = 4; off; off >>= 1)
            mm = fmaxf(mm, __shfl_xor(mm, off, 32));
        m = __shfl(mm, 0, 32);
    }
    __syncthreads();

    // block sum of exp
    const float e0 = __expf(x0 - m);
    const float e1 = __expf(x1 - m);
    float ssum = e0 + e1;
    for (int off = 16; off; off >>= 1)
        ssum += __shfl_xor(ssum, off, 32);
    if (lane == 0) red[wave] = ssum;
    __syncthreads();
    {
        float st = red[lane & 7];
        for (int off = 4; off; off >>= 1)
            st += __shfl_xor(st, off, 32);
        ssum = __shfl(st, 0, 32);
    }

    const float inv = 1.0f / ssum;
    out[(size_t)b * S_DIM + t]       = e0 * inv;
    out[(size_t)b * S_DIM + t + 256] = e1 * inv;
}

// ---------------------------------------------------------------------------
extern "C" void kernel_launch(void* const* d_in, const int* in_sizes, int n_in,
                              void* d_out, int out_size, void* d_ws, size_t ws_size,
                              hipStream_t stream) {
    const float* hidden = (const float*)d_in[0];   // [1,B,H]
    const float* enc    = (const float*)d_in[1];   // [S,B,H]
    const float* W      = (const float*)d_in[2];   // [H,H]
    const float* bias   = (const float*)d_in[3];   // [H]
    float* out = (float*)d_out;                    // [B,1,S]

    float* v      = (float*)d_ws;                  // B*H floats (512 KB)
    float* c      = v + B_DIM * H_DIM;             // B floats
    float* scores = c + 128;                       // B*S floats (256 KB), 16B-aligned

    gemm_v_kernel<<<dim3(B_DIM / 16, H_DIM / 16), 32, 0, stream>>>(hidden, W, v);
    bias_dot_kernel<<<B_DIM / 8, 256, 0, stream>>>(hidden, bias, c);
    scores_kernel<<<dim3(B_DIM, S_DIM / 64), 256, 0, stream>>>(enc, v, c, scores);
    softmax_kernel<<<B_DIM, 256, 0, stream>>>(scores, out);
}